// GCNLayer_23570780521023
// MI455X (gfx1250) — compile-verified
//
#include <hip/hip_runtime.h>

#define N_NODES 10000
#define IN_F    128
#define OUT_F   256

typedef __attribute__((ext_vector_type(2))) float v2f;
typedef __attribute__((ext_vector_type(8))) float v8f;

// ---------------------------------------------------------------------------
// Kernel 1: zero the h scratch (segment-sum accumulator), L2-resident.
// ---------------------------------------------------------------------------
__global__ void gcn_zero_ws(float* __restrict__ h, int n) {
    int i = blockIdx.x * blockDim.x + threadIdx.x;
    int stride = gridDim.x * blockDim.x;
    for (; i < n; i += stride) h[i] = 0.0f;
}

// ---------------------------------------------------------------------------
// Kernel 2: edge-parallel scatter-add (copy_u + sum).
// One wave32 per edge: 32 lanes x float4 = the whole 128-float feature row.
// Hardware f32 atomics (global_atomic_add_f32) resolve in L2 (h is 5.1MB).
// ---------------------------------------------------------------------------
__global__ __launch_bounds__(256) void gcn_scatter(
        const float* __restrict__ feat,
        const int*   __restrict__ src,
        const int*   __restrict__ dst,
        float*       __restrict__ h,
        int n_edges) {
    int edge = (blockIdx.x * blockDim.x + threadIdx.x) >> 5;
    int lane = threadIdx.x & 31;
    if (edge >= n_edges) return;
    int u = src[edge];
    int v = dst[edge];
    const float4* frow = (const float4*)(feat + (size_t)u * IN_F);
    float4 m = frow[lane];                       // coalesced b128 gather
    float* hp = h + (size_t)v * IN_F + lane * 4;
    unsafeAtomicAdd(hp + 0, m.x);
    unsafeAtomicAdd(hp + 1, m.y);
    unsafeAtomicAdd(hp + 2, m.z);
    unsafeAtomicAdd(hp + 3, m.w);
}

// ---------------------------------------------------------------------------
// Kernel 3: out = h @ W + b via V_WMMA_F32_16X16X4_F32 (full f32 precision).
// Block = 8 waves; block covers 128 rows x 16 cols; W slab staged in LDS.
// Per wave: one 16x16 tile, K=128 in 4 chunks of 8 WMMA steps.
// Loads for a whole chunk are issued before the chunk's WMMAs so the
// compiler can overlap memory latency (no per-step s_wait_loadcnt 0).
// NOTE: 10000 % 16 == 0 -> no M tail, so no lane-level guards anywhere.
// ---------------------------------------------------------------------------
__global__ __launch_bounds__(256) void gcn_gemm_wmma(
        const float* __restrict__ h,
        const float* __restrict__ W,
        const float* __restrict__ bias,
        float*       __restrict__ out) {
    __shared__ float Bs[IN_F * 16];              // W[:, col0:col0+16], 8 KB

    const int col0 = blockIdx.y * 16;

    // Cooperative load of the 128x16 W slab (reused by all 8 waves).
    for (int i = threadIdx.x; i < IN_F * 16; i += 256) {
        int kr = i >> 4;
        int c  = i & 15;
        Bs[i] = W[kr * OUT_F + col0 + c];
    }
    __syncthreads();

    const int wave   = threadIdx.x >> 5;
    const int lane   = threadIdx.x & 31;
    const int m_tile = blockIdx.x * 8 + wave;
    if (m_tile >= N_NODES / 16) return;          // wave-uniform exit, no tail tiles
    const int row0 = m_tile * 16;

    const int half = lane >> 4;                  // 0: lanes 0-15, 1: lanes 16-31
    const int l16  = lane & 15;

    // A-fragment layout (32-bit A 16x4): lanes 0-15 -> K = k,k+1 ;
    //                                    lanes 16-31 -> K = k+2,k+3 ; M = l16.
    const float* aptr = h + (size_t)(row0 + l16) * IN_F + half * 2;
    // B-fragment base: row kb striped across lanes (kb = k + 2*half).
    const int bbase = half * 2 * 16 + l16;

    v8f acc = {};
    #pragma unroll
    for (int kk = 0; kk < IN_F; kk += 32) {      // 4 chunks of 8 WMMA steps
        v2f a[8], bf[8];
        #pragma unroll
        for (int s = 0; s < 8; ++s) {            // batch all loads first
            const int k = kk + s * 4;
            a[s] = *(const v2f*)(aptr + k);      // global b64, no guard
            bf[s].x = Bs[k * 16 + bbase];        // ds_load_2addr pair
            bf[s].y = Bs[k * 16 + bbase + 16];
        }
        #pragma unroll
        for (int s = 0; s < 8; ++s) {
            // 8-arg form: (neg_a, A, neg_b, B, c_mod, C, reuse_a, reuse_b)
            acc = __builtin_amdgcn_wmma_f32_16x16x4_f32(
                    false, a[s], false, bf[s], (short)0, acc, false, false);
        }
    }

    // C/D layout: VGPR j -> (M = j,    N = lane)     for lanes 0-15,
    //                       (M = j+8,  N = lane-16)  for lanes 16-31.
    const float bv = bias[col0 + l16];
    #pragma unroll
    for (int j = 0; j < 8; ++j) {
        int r = row0 + j + half * 8;
        out[(size_t)r * OUT_F + col0 + l16] = acc[j] + bv;
    }
}

// ---------------------------------------------------------------------------
extern "C" void kernel_launch(void* const* d_in, const int* in_sizes, int n_in,
                              void* d_out, int out_size, void* d_ws, size_t ws_size,
                              hipStream_t stream) {
    const float* feat = (const float*)d_in[0];   // [10000, 128] f32
    const int*   src  = (const int*)  d_in[1];   // [E] i32
    const int*   dst  = (const int*)  d_in[2];   // [E] i32
    const float* W    = (const float*)d_in[3];   // [128, 256] f32
    const float* bias = (const float*)d_in[4];   // [256] f32
    float*       out  = (float*)      d_out;     // [10000, 256] f32
    float*       h    = (float*)      d_ws;      // [10000, 128] f32 scratch

    const int n_edges = in_sizes[1];
    const int h_elems = N_NODES * IN_F;

    gcn_zero_ws<<<(h_elems + 255) / 256, 256, 0, stream>>>(h, h_elems);

    gcn_scatter<<<(n_edges + 7) / 8, 256, 0, stream>>>(feat, src, dst, h, n_edges);

    const int m_tiles = N_NODES / 16;                // 625, exact
    dim3 grid((m_tiles + 7) / 8, OUT_F / 16);        // (79, 16)
    gcn_gemm_wmma<<<grid, 256, 0, stream>>>(h, W, bias, out);
}